// BPNet_61778809585864
// MI455X (gfx1250) — compile-verified
//
#include <hip/hip_runtime.h>
#include <math.h>

// ---------------------------------------------------------------------------
// CDNA5 (gfx1250, wave32) implementation of the BPNet/DGCNN + Sinkhorn ref.
// GEMMs (edge convs, 512-d projection, 2048x2048 score matrix) use
// V_WMMA_F32_16X16X4_F32 with 2x2 register blocking (32x32 macro-tile per
// wave): 4 WMMAs share each A/B fragment load -> 2x arithmetic intensity.
// Sinkhorn uses single-pass online logsumexp (bandwidth-bound: ~13 GB total).
// ---------------------------------------------------------------------------

typedef __attribute__((ext_vector_type(2))) float v2f;
typedef __attribute__((ext_vector_type(8))) float v8f;

#define BATCH   8
#define NPTS    2048
#define K_NN    20
#define PADN    (NPTS + 1)      // 2049
#define EPSF    1e-5f
#define LAMBF   0.5f
#define OUTLF   0.01f
#define NITERS  50

#define WMMA_F32(A, B, Cacc) \
  __builtin_amdgcn_wmma_f32_16x16x4_f32(false, (A), false, (B), (short)0, (Cacc), false, false)

// ------------------------------- utility -----------------------------------

__global__ void k_zero(float* __restrict__ p, size_t n) {
  size_t i = (size_t)blockIdx.x * blockDim.x + threadIdx.x;
  if (i < n) p[i] = 0.f;
}

// --------------------- input matchnorm (B,3,N), 3D path ---------------------
__global__ void k_matchnorm_in(const float* __restrict__ x, float* __restrict__ xn,
                               float* __restrict__ scale, int use_stored) {
  __shared__ float red[256];
  int b = blockIdx.x, t = threadIdx.x;
  const float* xb = x + (size_t)b * 3 * NPTS;
  if (!use_stored) {
    float mx = 0.f;
    for (int i = t; i < 3 * NPTS; i += 256) mx = fmaxf(mx, fabsf(xb[i]));
    red[t] = mx; __syncthreads();
    for (int o = 128; o > 0; o >>= 1) {
      if (t < o) red[t] = fmaxf(red[t], red[t + o]);
      __syncthreads();
    }
    if (t == 0) scale[b] = red[0];
    __syncthreads();
  }
  float inv = 1.f / scale[b];
  for (int c = 0; c < 3; c++) {
    float sum = 0.f;
    for (int i = t; i < NPTS; i += 256) sum += xb[c * NPTS + i] * inv;
    red[t] = sum; __syncthreads();
    for (int o = 128; o > 0; o >>= 1) {
      if (t < o) red[t] += red[t + o];
      __syncthreads();
    }
    float mean = red[0] / (float)NPTS;
    __syncthreads();
    for (int i = t; i < NPTS; i += 256)
      xn[(size_t)b * 3 * NPTS + c * NPTS + i] = xb[c * NPTS + i] * inv - mean;
    __syncthreads();
  }
}

// ------------------------- kNN (top-K of -||xi-xj||^2) ----------------------
__global__ void k_knn(const float* __restrict__ xn, int* __restrict__ idx) {
  __shared__ float px[NPTS], py[NPTS], pz[NPTS];
  int b = blockIdx.x >> 3;
  int n0 = (blockIdx.x & 7) * 256;
  const float* xb = xn + (size_t)b * 3 * NPTS;
  for (int i = threadIdx.x; i < NPTS; i += 256) {
    px[i] = xb[i]; py[i] = xb[NPTS + i]; pz[i] = xb[2 * NPTS + i];
  }
  __syncthreads();
  int n = n0 + threadIdx.x;
  float qx = px[n], qy = py[n], qz = pz[n];
  float sqn = qx * qx + qy * qy + qz * qz;
  float bestd[K_NN]; int besti[K_NN];
#pragma unroll
  for (int k = 0; k < K_NN; k++) { bestd[k] = -3.4e38f; besti[k] = 0; }
  for (int m = 0; m < NPTS; m++) {
    float dx = px[m], dy = py[m], dz = pz[m];
    float d = 2.f * (qx * dx + qy * dy + qz * dz) - sqn - (dx * dx + dy * dy + dz * dz);
    if (d > bestd[K_NN - 1]) {
      int pos = K_NN - 1;
      while (pos > 0 && bestd[pos - 1] < d) {
        bestd[pos] = bestd[pos - 1]; besti[pos] = besti[pos - 1]; pos--;
      }
      bestd[pos] = d; besti[pos] = m;
    }
  }
  int* ob = idx + ((size_t)b * NPTS + n) * K_NN;
#pragma unroll
  for (int k = 0; k < K_NN; k++) ob[k] = besti[k];
}

// ------------------- edge features feat[b,6,n,k] -----------------------------
__global__ void k_edge(const float* __restrict__ xn, const int* __restrict__ idx,
                       float* __restrict__ feat) {
  int i = blockIdx.x * blockDim.x + threadIdx.x;
  const int total = BATCH * NPTS * K_NN;
  if (i >= total) return;
  int kk = i % K_NN;
  int n = (i / K_NN) % NPTS;
  int b = i / (K_NN * NPTS);
  int j = idx[i];
  const float* xb = xn + (size_t)b * 3 * NPTS;
  const size_t P = (size_t)NPTS * K_NN;
  float* fb = feat + (size_t)b * 6 * P + (size_t)n * K_NN + kk;
  fb[0 * P] = xb[j];            fb[1 * P] = xb[NPTS + j];     fb[2 * P] = xb[2 * NPTS + j];
  fb[3 * P] = xb[n];            fb[4 * P] = xb[NPTS + n];     fb[5 * P] = xb[2 * NPTS + n];
}

// ---- GEMM via V_WMMA_F32_16X16X4_F32, 32x32 macro-tile (2x2 blocking) ------
// y[b,o,p] = sum_c w[o,c] * x[b,c,p];  O,P multiples of 32.
__global__ void k_conv_wmma(const float* __restrict__ w, const float* __restrict__ x,
                            float* __restrict__ y, int O, int C, int P) {
  int wave = blockIdx.x * (blockDim.x >> 5) + (threadIdx.x >> 5);
  int lane = threadIdx.x & 31;
  int tP = P >> 5, tO = O >> 5;
  int perB = tO * tP;
  int b = wave / perB;
  if (b >= BATCH) return;                 // uniform per wave; EXEC stays full
  int r = wave % perB;
  int ot = (r / tP) * 32, pt = (r % tP) * 32;
  int m16 = lane & 15;                    // A-row (M) / B-col (N) within tile
  int kh = lane >> 4;                     // which K-pair this half holds
  const float* w0 = w + (size_t)(ot + m16) * C;
  const float* w1 = w0 + (size_t)16 * C;
  const float* xb = x + (size_t)b * C * P + pt + m16;
  v8f a00 = {}, a01 = {}, a10 = {}, a11 = {};
  for (int k0 = 0; k0 < C; k0 += 4) {
    int ka = k0 + 2 * kh;
    bool u0 = ka < C, u1 = (ka + 1) < C;
    v2f A0, A1, B0, B1;
    A0.x = u0 ? w0[ka]     : 0.f;  A0.y = u1 ? w0[ka + 1] : 0.f;
    A1.x = u0 ? w1[ka]     : 0.f;  A1.y = u1 ? w1[ka + 1] : 0.f;
    B0.x = u0 ? xb[(size_t)ka * P]      : 0.f;
    B0.y = u1 ? xb[(size_t)(ka + 1) * P] : 0.f;
    B1.x = u0 ? xb[(size_t)ka * P + 16]      : 0.f;
    B1.y = u1 ? xb[(size_t)(ka + 1) * P + 16] : 0.f;
    a00 = WMMA_F32(A0, B0, a00);
    a01 = WMMA_F32(A0, B1, a01);
    a10 = WMMA_F32(A1, B0, a10);
    a11 = WMMA_F32(A1, B1, a11);
  }
  int nn = lane & 15, mh = lane >> 4;
  float* yb = y + (size_t)b * O * P + (size_t)ot * P + pt + nn;
#pragma unroll
  for (int rr = 0; rr < 8; rr++) {
    size_t row = (size_t)(rr + 8 * mh);
    yb[row * P]             = a00[rr];
    yb[row * P + 16]        = a01[rr];
    yb[(row + 16) * P]      = a10[rr];
    yb[(row + 16) * P + 16] = a11[rr];
  }
}

// ------------------ per-batch max|y| (float bits atomicMax) ------------------
__global__ void k_absmax(const float* __restrict__ y, float* __restrict__ scale,
                         size_t perBatch) {
  __shared__ float red[256];
  int bpb = gridDim.x / BATCH;
  int b = blockIdx.x / bpb;
  int blk = blockIdx.x % bpb;
  const float* yb = y + (size_t)b * perBatch;
  float mx = 0.f;
  for (size_t i = (size_t)blk * blockDim.x + threadIdx.x; i < perBatch;
       i += (size_t)bpb * blockDim.x)
    mx = fmaxf(mx, fabsf(yb[i]));
  red[threadIdx.x] = mx; __syncthreads();
  for (int o = 128; o > 0; o >>= 1) {
    if (threadIdx.x < o) red[threadIdx.x] = fmaxf(red[threadIdx.x], red[threadIdx.x + o]);
    __syncthreads();
  }
  if (threadIdx.x == 0)
    atomicMax((unsigned int*)&scale[b], __float_as_uint(red[0]));
}

// ---- matchnorm 4D: divide by scale, subtract mean of each 2048-elem chunk ---
__global__ void k_mn_apply(float* __restrict__ y, const float* __restrict__ scale) {
  __shared__ float red[256];
  size_t chunk = blockIdx.x;
  size_t chunksPerB = gridDim.x / BATCH;
  int b = (int)(chunk / chunksPerB);
  float inv = 1.f / scale[b];
  float* yc = y + chunk * NPTS;
  float sum = 0.f;
  for (int i = threadIdx.x; i < NPTS; i += 256) sum += yc[i] * inv;
  red[threadIdx.x] = sum; __syncthreads();
  for (int o = 128; o > 0; o >>= 1) {
    if (threadIdx.x < o) red[threadIdx.x] += red[threadIdx.x + o];
    __syncthreads();
  }
  float mean = red[0] / (float)NPTS;
  for (int i = threadIdx.x; i < NPTS; i += 256) yc[i] = yc[i] * inv - mean;
}

// -------------------- batchnorm stats per channel ---------------------------
__global__ void k_bnstats(const float* __restrict__ y, float* __restrict__ mv,
                          int O, size_t P) {
  __shared__ float rs[256], rq[256];
  int o = blockIdx.x;
  float s = 0.f, q = 0.f;
  for (int b = 0; b < BATCH; b++) {
    const float* yb = y + ((size_t)b * O + o) * P;
    for (size_t i = threadIdx.x; i < P; i += 256) { float v = yb[i]; s += v; q += v * v; }
  }
  rs[threadIdx.x] = s; rq[threadIdx.x] = q; __syncthreads();
  for (int off = 128; off > 0; off >>= 1) {
    if (threadIdx.x < off) {
      rs[threadIdx.x] += rs[threadIdx.x + off];
      rq[threadIdx.x] += rq[threadIdx.x + off];
    }
    __syncthreads();
  }
  if (threadIdx.x == 0) {
    float cnt = (float)((size_t)BATCH * P);
    float mean = rs[0] / cnt;
    mv[o] = mean;
    mv[O + o] = rq[0] / cnt - mean * mean;
  }
}

__global__ void k_bnrelu(float* __restrict__ y, const float* __restrict__ mv,
                         const float* __restrict__ gamma, const float* __restrict__ beta,
                         int O, size_t P) {
  size_t i = (size_t)blockIdx.x * blockDim.x + threadIdx.x;
  size_t total = (size_t)BATCH * O * P;
  if (i >= total) return;
  int o = (int)((i / P) % O);
  float v = (y[i] - mv[o]) * rsqrtf(mv[O + o] + EPSF) * gamma[o] + beta[o];
  y[i] = fmaxf(v, 0.f);
}

// -------------------------- max over k neighbors ----------------------------
__global__ void k_maxk(const float* __restrict__ y, float* __restrict__ xc,
                       int O, int chOff) {
  int i = blockIdx.x * blockDim.x + threadIdx.x;
  int total = BATCH * O * NPTS;
  if (i >= total) return;
  int n = i % NPTS;
  int o = (i / NPTS) % O;
  int b = i / (NPTS * O);
  const float* yb = y + (((size_t)b * O + o) * NPTS + n) * K_NN;
  float m = yb[0];
#pragma unroll
  for (int k = 1; k < K_NN; k++) m = fmaxf(m, yb[k]);
  xc[((size_t)b * 512 + chOff + o) * NPTS + n] = m;
}

// ------ scores: s[b,n,m] = (es^T et)/(sqrt(512)*LAMB), 32x32 macro-tiles -----
__global__ void k_scores(const float* __restrict__ es, const float* __restrict__ et,
                         float* __restrict__ s) {
  int wave = blockIdx.x * (blockDim.x >> 5) + (threadIdx.x >> 5);
  int lane = threadIdx.x & 31;
  const int t32 = NPTS / 32;              // 64 macro-tiles per dim
  const int perB = t32 * t32;
  int b = wave / perB;
  if (b >= BATCH) return;
  int r = wave % perB;
  int nt = (r / t32) * 32, mt = (r % t32) * 32;
  int m16 = lane & 15, kh = lane >> 4;
  const float* eA0 = es + (size_t)b * 512 * NPTS + nt + m16;
  const float* eA1 = eA0 + 16;
  const float* eB0 = et + (size_t)b * 512 * NPTS + mt + m16;
  const float* eB1 = eB0 + 16;
  v8f a00 = {}, a01 = {}, a10 = {}, a11 = {};
  for (int k = 0; k < 512; k += 4) {
    size_t ka = (size_t)(k + 2 * kh);
    v2f A0, A1, B0, B1;
    A0.x = eA0[ka * NPTS];  A0.y = eA0[(ka + 1) * NPTS];
    A1.x = eA1[ka * NPTS];  A1.y = eA1[(ka + 1) * NPTS];
    B0.x = eB0[ka * NPTS];  B0.y = eB0[(ka + 1) * NPTS];
    B1.x = eB1[ka * NPTS];  B1.y = eB1[(ka + 1) * NPTS];
    a00 = WMMA_F32(A0, B0, a00);
    a01 = WMMA_F32(A0, B1, a01);
    a10 = WMMA_F32(A1, B0, a10);
    a11 = WMMA_F32(A1, B1, a11);
  }
  const float scl = (1.f / LAMBF) * 0.04419417382415922f; // 1/sqrt(512)
  int nn = lane & 15, mh = lane >> 4;
  float* sb = s + (size_t)b * PADN * PADN + (size_t)nt * PADN + mt + nn;
#pragma unroll
  for (int rr = 0; rr < 8; rr++) {
    size_t row = (size_t)(rr + 8 * mh);
    sb[row * PADN]             = a00[rr] * scl;
    sb[row * PADN + 16]        = a01[rr] * scl;
    sb[(row + 16) * PADN]      = a10[rr] * scl;
    sb[(row + 16) * PADN + 16] = a11[rr] * scl;
  }
}

__global__ void k_pad(float* __restrict__ s) {
  int i = blockIdx.x * blockDim.x + threadIdx.x;
  if (i >= BATCH * PADN) return;
  int b = i / PADN, j = i % PADN;
  float* sb = s + (size_t)b * PADN * PADN;
  sb[(size_t)NPTS * PADN + j] = OUTLF;    // last row
  sb[(size_t)j * PADN + NPTS] = OUTLF;    // last col
}

// ------------- sinkhorn: online logsumexp, row (f) and column (g) -----------
__global__ void k_sink_f(const float* __restrict__ s, const float* __restrict__ g,
                         float* __restrict__ f) {
  __shared__ float rm[256], rs[256];
  int b = blockIdx.x / PADN;
  int i = blockIdx.x % PADN;
  const float* srow = s + ((size_t)b * PADN + i) * PADN;
  const float* gb = g + (size_t)b * PADN;
  float m = -3.4e38f, acc = 0.f;
  for (int j = threadIdx.x; j < PADN; j += 256) {
    float v = srow[j] + gb[j];
    float M = fmaxf(m, v);
    acc = acc * __expf(m - M) + __expf(v - M);
    m = M;
  }
  rm[threadIdx.x] = m; rs[threadIdx.x] = acc; __syncthreads();
  for (int o = 128; o > 0; o >>= 1) {
    if (threadIdx.x < o) {
      float M = fmaxf(rm[threadIdx.x], rm[threadIdx.x + o]);
      rs[threadIdx.x] = rs[threadIdx.x] * __expf(rm[threadIdx.x] - M)
                      + rs[threadIdx.x + o] * __expf(rm[threadIdx.x + o] - M);
      rm[threadIdx.x] = M;
    }
    __syncthreads();
  }
  if (threadIdx.x == 0) {
    float fa = (i == NPTS) ? (LAMBF * logf((float)NPTS)) : 0.f;
    f[(size_t)b * PADN + i] = fa - (rm[0] + logf(rs[0]));
  }
}

__global__ void k_sink_g(const float* __restrict__ s, const float* __restrict__ f,
                         float* __restrict__ g) {
  int t = blockIdx.x * blockDim.x + threadIdx.x;
  if (t >= BATCH * PADN) return;
  int b = t / PADN, j = t % PADN;
  const float* sb = s + (size_t)b * PADN * PADN + j;
  const float* fb = f + (size_t)b * PADN;
  float m = -3.4e38f, acc = 0.f;
  for (int i = 0; i < PADN; i++) {                 // coalesced across lanes
    float v = sb[(size_t)i * PADN] + fb[i];
    float M = fmaxf(m, v);
    acc = acc * __expf(m - M) + __expf(v - M);
    m = M;
  }
  float fbv = (j == NPTS) ? (LAMBF * logf((float)NPTS)) : 0.f;
  g[t] = fbv - (m + logf(acc));
}

__global__ void k_sink_final(float* __restrict__ s, const float* __restrict__ f,
                             const float* __restrict__ g) {
  size_t i = (size_t)blockIdx.x * blockDim.x + threadIdx.x;
  const size_t per = (size_t)PADN * PADN;
  const size_t total = (size_t)BATCH * per;
  if (i >= total) return;
  int b = (int)(i / per);
  size_t r = i % per;
  int ii = (int)(r / PADN), jj = (int)(r % PADN);
  s[i] = __expf(s[i] + f[(size_t)b * PADN + ii] + g[(size_t)b * PADN + jj]);
}

// ------------------------------ host side -----------------------------------

static void conv_block(const float* w, const float* gam, const float* bet,
                       const float* xin, float* yout, int O, int C, int P,
                       float* scaleSlot, float* bn_mv, int first, hipStream_t st) {
  int tiles = BATCH * (O / 32) * (P / 32);
  k_conv_wmma<<<(tiles + 3) / 4, 128, 0, st>>>(w, xin, yout, O, C, P);
  size_t perBatch = (size_t)O * P;
  if (first) {
    k_zero<<<1, 64, 0, st>>>(scaleSlot, BATCH);
    k_absmax<<<BATCH * 256, 256, 0, st>>>(yout, scaleSlot, perBatch);
  }
  int chunks = (int)((size_t)BATCH * perBatch / NPTS);
  k_mn_apply<<<chunks, 256, 0, st>>>(yout, scaleSlot);
  k_bnstats<<<O, 256, 0, st>>>(yout, bn_mv, O, (size_t)P);
  size_t total = (size_t)BATCH * perBatch;
  k_bnrelu<<<(int)((total + 255) / 256), 256, 0, st>>>(yout, bn_mv, gam, bet, O, (size_t)P);
}

extern "C" void kernel_launch(void* const* d_in, const int* in_sizes, int n_in,
                              void* d_out, int out_size, void* d_ws, size_t ws_size,
                              hipStream_t stream) {
  (void)in_sizes; (void)n_in; (void)out_size; (void)ws_size;
  const float* src = (const float*)d_in[0];
  const float* tgt = (const float*)d_in[1];
  // params flattened in insertion order: w1,g1,b1, w2,g2,b2, ... w5,g5,b5
  const float* W[5]; const float* G[5]; const float* Bt[5];
  for (int i = 0; i < 5; i++) {
    W[i]  = (const float*)d_in[2 + 3 * i];
    G[i]  = (const float*)d_in[3 + 3 * i];
    Bt[i] = (const float*)d_in[4 + 3 * i];
  }
  const int OArr[5] = {64, 64, 128, 256, 512};
  const int CArr[5] = {6, 64, 64, 128, 512};

  // ---------------- workspace layout (floats) ----------------
  float* scales = (float*)d_ws;                   // 6 slots x 8 batches (64)
  float* bn_mv  = scales + 64;                    // 1024
  float* fvec   = bn_mv + 1024;                   // 8*2049
  float* gvec   = fvec + BATCH * PADN;            // 8*2049
  float* xn     = gvec + BATCH * PADN;            // 8*3*2048
  int*   idx    = (int*)(xn + (size_t)BATCH * 3 * NPTS);          // 8*2048*20
  float* feat   = (float*)(idx + (size_t)BATCH * NPTS * K_NN);    // 8*6*40960
  float* bufA   = feat + (size_t)BATCH * 6 * NPTS * K_NN;         // 8*128*40960
  float* bufB   = bufA + (size_t)BATCH * 128 * NPTS * K_NN;       // 8*256*40960
  float* xc     = bufB + (size_t)BATCH * 256 * NPTS * K_NN;       // 8*512*2048
  float* embS   = xc + (size_t)BATCH * 512 * NPTS;                // 8*512*2048
  float* embT   = embS + (size_t)BATCH * 512 * NPTS;              // 8*512*2048

  const int P = NPTS * K_NN;  // 40960

  for (int pass = 0; pass < 2; pass++) {
    int first = (pass == 0) ? 1 : 0;
    const float* x = first ? src : tgt;
    float* emb = first ? embS : embT;
    k_matchnorm_in<<<BATCH, 256, 0, stream>>>(x, xn, scales + 0, first ? 0 : 1);
    k_knn<<<BATCH * 8, 256, 0, stream>>>(xn, idx);
    k_edge<<<(BATCH * NPTS * K_NN + 255) / 256, 256, 0, stream>>>(xn, idx, feat);

    conv_block(W[0], G[0], Bt[0], feat, bufA, OArr[0], CArr[0], P, scales + 8,  bn_mv, first, stream);
    k_maxk<<<(BATCH * 64 * NPTS + 255) / 256, 256, 0, stream>>>(bufA, xc, 64, 0);
    conv_block(W[1], G[1], Bt[1], bufA, bufB, OArr[1], CArr[1], P, scales + 16, bn_mv, first, stream);
    k_maxk<<<(BATCH * 64 * NPTS + 255) / 256, 256, 0, stream>>>(bufB, xc, 64, 64);
    conv_block(W[2], G[2], Bt[2], bufB, bufA, OArr[2], CArr[2], P, scales + 24, bn_mv, first, stream);
    k_maxk<<<(BATCH * 128 * NPTS + 255) / 256, 256, 0, stream>>>(bufA, xc, 128, 128);
    conv_block(W[3], G[3], Bt[3], bufA, bufB, OArr[3], CArr[3], P, scales + 32, bn_mv, first, stream);
    k_maxk<<<(BATCH * 256 * NPTS + 255) / 256, 256, 0, stream>>>(bufB, xc, 256, 256);
    conv_block(W[4], G[4], Bt[4], xc, emb, OArr[4], CArr[4], NPTS, scales + 40, bn_mv, first, stream);
    // reference applies relu(emb) again -- idempotent after bn+relu.
  }

  // scores -> padded s stored directly in d_out
  float* s = (float*)d_out;
  int scoreTiles = BATCH * (NPTS / 32) * (NPTS / 32);
  k_scores<<<(scoreTiles + 3) / 4, 128, 0, stream>>>(embS, embT, s);
  k_pad<<<(BATCH * PADN + 255) / 256, 256, 0, stream>>>(s);

  // sinkhorn
  k_zero<<<(2 * BATCH * PADN + 255) / 256, 256, 0, stream>>>(fvec, (size_t)2 * BATCH * PADN);
  for (int it = 0; it < NITERS; it++) {
    k_sink_f<<<BATCH * PADN, 256, 0, stream>>>(s, gvec, fvec);
    k_sink_g<<<(BATCH * PADN + 255) / 256, 256, 0, stream>>>(s, fvec, gvec);
  }
  size_t total = (size_t)BATCH * PADN * PADN;
  k_sink_final<<<(int)((total + 255) / 256), 256, 0, stream>>>(s, fvec, gvec);
}